// RelationalNet_21199958573582
// MI455X (gfx1250) — compile-verified
//
#include <hip/hip_runtime.h>
#include <hip/hip_bf16.h>

// RelationalNet on MI455X (gfx1250, wave32, WMMA).
// B=32, L1=L2=64, H=256, FEAT=512.
//
// Phase 1: y1 = x1 @ W[:, :F]^T, y2 = x2 @ W[:, F:]^T via V_WMMA_F32_16X16X4_F32
// Phase 2: per (b, i-chunk) register-blocked relu-pair reduction (VALU, L2-resident)
// Phase 3: fold partials + padding term relu(bias)*(L1*L2 - s1*s2), divide by s1*s2

typedef __attribute__((ext_vector_type(2))) float v2f;
typedef __attribute__((ext_vector_type(8))) float v8f;

#define BB   32
#define LL   64
#define HH   256
#define FF   512

// ws layout (floats)
#define Y1_OFF   0
#define Y2_OFF   (BB * LL * HH)              // 524288
#define PART_OFF (2 * BB * LL * HH)          // 1048576
#define NCHUNK   4                           // i-chunks of 16 rows

// ---------------------------------------------------------------------------
// Phase 1: one wave per 16x16 (i x h) output tile.
// tiles = 2 gemms * 32 batches * 4 iTiles * 16 hTiles = 4096 waves
// A frag (16x4 f32, 2 VGPR): lanes 0-15 -> K={k0,k0+1}, lanes 16-31 -> K={k0+2,k0+3}
// B frag (4x16 f32, 2 VGPR): same half-wave K split, N = lane&15
// D (16x16 f32, 8 VGPR): VGPR v -> M = v + 8*(lane>=16), N = lane&15
// ---------------------------------------------------------------------------
__global__ void __launch_bounds__(256)
relnet_gemm_wmma(const float* __restrict__ x1, const float* __restrict__ x2,
                 const float* __restrict__ W, float* __restrict__ ws) {
    const int wave  = (blockIdx.x * blockDim.x + threadIdx.x) >> 5;  // 0..4095
    const int lane  = threadIdx.x & 31;

    const int hTile = wave & 15;          // 16 h-tiles
    const int iTile = (wave >> 4) & 3;    // 4 i-tiles
    const int batch = (wave >> 6) & 31;   // 32 batches
    const int g     = wave >> 11;         // 0: x1/W-left, 1: x2/W-right

    const float* __restrict__ X = g ? x2 : x1;
    float*       __restrict__ Y = ws + (g ? Y2_OFF : Y1_OFF);
    const int    wcol = g ? FF : 0;       // column offset into W rows (width 2F)

    const int i0 = iTile * 16;
    const int h0 = hTile * 16;
    const int lm = lane & 15;
    const int hs = lane >> 4;             // half-wave select: K base 0 or 2

    // A: row (i0+lm) of X[batch], two consecutive K at (k0 + 2*hs)
    const float* aptr = X + ((size_t)batch * LL + (i0 + lm)) * FF + 2 * hs;
    // B: B[k][n] = W[h0+n][wcol+k]; lane holds n=lm, two consecutive K at (k0 + 2*hs)
    const float* bptr = W + (size_t)(h0 + lm) * (2 * FF) + wcol + 2 * hs;

    v8f c = {0.f, 0.f, 0.f, 0.f, 0.f, 0.f, 0.f, 0.f};

    #pragma unroll 8
    for (int k0 = 0; k0 < FF; k0 += 4) {
        v2f a = *(const v2f*)(aptr + k0);   // 8B-aligned
        v2f bfrag = *(const v2f*)(bptr + k0);
        // (neg_a, A, neg_b, B, c_mod, C, reuse_a, reuse_b)
        c = __builtin_amdgcn_wmma_f32_16x16x4_f32(
                false, a, false, bfrag, (short)0, c, false, false);
    }

    float* yout = Y + ((size_t)batch * LL + i0 + 8 * hs) * HH + h0 + lm;
    #pragma unroll
    for (int v = 0; v < 8; ++v) {
        yout[(size_t)v * HH] = c[v];
    }
}

// ---------------------------------------------------------------------------
// Phase 2: block = (batch, 16-row i-chunk), 256 threads = one h each.
// v1[ii] = y1[b, i0+ii, h] + bias[h]  (or -inf when i >= s1 so relu yields 0)
// stream y2[b, j, h] once per j, 16 add+max per load. All L2-resident.
// ---------------------------------------------------------------------------
__global__ void __launch_bounds__(256)
relnet_pairsum(const float* __restrict__ ws_y, const float* __restrict__ bias,
               const int* __restrict__ sl1, const int* __restrict__ sl2,
               float* __restrict__ partial) {
    const int b     = blockIdx.x >> 2;   // 0..31
    const int chunk = blockIdx.x & 3;    // 0..3
    const int h     = threadIdx.x;       // 0..255

    const int s1 = sl1[b];
    const int s2 = sl2[b];
    const float bh = bias[h];

    const float* __restrict__ y1b = ws_y + Y1_OFF + (size_t)b * LL * HH + h;
    const float* __restrict__ y2b = ws_y + Y2_OFF + (size_t)b * LL * HH + h;

    const int i0 = chunk * 16;
    float v1[16];
    #pragma unroll
    for (int ii = 0; ii < 16; ++ii) {
        const int i = i0 + ii;
        v1[ii] = (i < s1) ? (y1b[(size_t)i * HH] + bh) : -__builtin_inff();
    }

    float acc = 0.f;
    for (int j = 0; j < s2; ++j) {
        const float v2 = y2b[(size_t)j * HH];
        #pragma unroll
        for (int ii = 0; ii < 16; ++ii) {
            acc += fmaxf(0.f, v1[ii] + v2);
        }
    }

    partial[((size_t)b * NCHUNK + chunk) * HH + h] = acc;
}

// ---------------------------------------------------------------------------
// Phase 3: out[b,h] = (sum_partials + (L1*L2 - s1*s2)*relu(bias[h])) / (s1*s2)
// ---------------------------------------------------------------------------
__global__ void __launch_bounds__(256)
relnet_finalize(const float* __restrict__ partial, const float* __restrict__ bias,
                const int* __restrict__ sl1, const int* __restrict__ sl2,
                float* __restrict__ out) {
    const int b = blockIdx.x;
    const int h = threadIdx.x;
    const int s1 = sl1[b];
    const int s2 = sl2[b];

    float sum = 0.f;
    #pragma unroll
    for (int c = 0; c < NCHUNK; ++c) {
        sum += partial[((size_t)b * NCHUNK + c) * HH + h];
    }
    const float pad = (float)(LL * LL - s1 * s2) * fmaxf(0.f, bias[h]);
    out[(size_t)b * HH + h] = (sum + pad) / (float)(s1 * s2);
}

extern "C" void kernel_launch(void* const* d_in, const int* in_sizes, int n_in,
                              void* d_out, int out_size, void* d_ws, size_t ws_size,
                              hipStream_t stream) {
    const float* x1   = (const float*)d_in[0];
    const int*   sl1  = (const int*)  d_in[1];
    const float* x2   = (const float*)d_in[2];
    const int*   sl2  = (const int*)  d_in[3];
    const float* W    = (const float*)d_in[4];
    const float* bias = (const float*)d_in[5];
    float*       out  = (float*)d_out;
    float*       ws   = (float*)d_ws;

    (void)in_sizes; (void)n_in; (void)out_size; (void)ws_size;

    // 4096 tile-waves, 8 waves (256 threads) per block -> 512 blocks
    relnet_gemm_wmma<<<512, 256, 0, stream>>>(x1, x2, W, ws);
    // 32 batches * 4 i-chunks
    relnet_pairsum<<<BB * NCHUNK, HH, 0, stream>>>(ws, bias, sl1, sl2, ws + PART_OFF);
    relnet_finalize<<<BB, HH, 0, stream>>>(ws + PART_OFF, bias, sl1, sl2, out);
}